// ds_us_89472758710788
// MI455X (gfx1250) — compile-verified
//
#include <hip/hip_runtime.h>
#include <hip/hip_bf16.h>
#include <stdint.h>

// Problem constants (from reference): B=32, C=256. N_in, N_out, NNZ derived
// host-side from in_sizes/out_size so the launch stays deterministic.
#define BATCH     32
#define CFEAT     256
#define CW        32          // channel columns per workgroup (== wave width)
#define NTHREADS  512         // 16 waves (wave32)
#define NWAVES    (NTHREADS / 32)
#define CHUNK     512         // COO entries staged per TDM chunk
#define SUB       (CHUNK / NWAVES)   // 32 entries per wave per chunk

typedef __attribute__((ext_vector_type(4))) float        v4f;
typedef __attribute__((ext_vector_type(4))) int          v4i;
typedef __attribute__((ext_vector_type(4))) unsigned int v4u;
typedef __attribute__((ext_vector_type(8))) unsigned int v8u;

// Issue one TDM descriptor: 1D copy of `CHUNK` 4-byte elements from gptr to
// LDS offset lds_off; elements past `valid_elems` read as zero (HW OOB rule).
// D# layout per CDNA5 ISA ch.8.3/8.4. TENSORcnt-tracked.
__device__ __forceinline__ void tdm_load_1d(const void* gptr, uint32_t lds_off,
                                            uint32_t valid_elems) {
    const unsigned long long ga = (unsigned long long)(size_t)gptr;
    v4u g0;
    g0[0] = 0x1u;                                   // count=1, user descriptor
    g0[1] = lds_off;                                // lds_addr (bytes)
    g0[2] = (uint32_t)ga;                           // global_addr[31:0]
    g0[3] = (uint32_t)((ga >> 32) & 0x01FFFFFFu)    // global_addr[56:32]
          | 0x80000000u;                            // type=2 ("image")
    v8u g1;
    g1[0] = 0x00020000u;                            // data_size=2 (4B), no mask
    g1[1] = (valid_elems & 0xFFFFu) << 16;          // tensor_dim0[15:0]
    g1[2] = (valid_elems >> 16) & 0xFFFFu;          // tensor_dim0[31:16]
    g1[3] = ((uint32_t)CHUNK) << 16;                // tile_dim0 = 512
    g1[4] = 0u;                                     // tile_dim1/2 = 0 (1D)
    g1[5] = (uint32_t)CHUNK;                        // dim0 stride (unused-safe)
    g1[6] = 0u;
    g1[7] = 0u;
    const v4u gz = (v4u)0u;   // zeroed groups 2/3: dims 2-4 / iterate disabled
    asm volatile("tensor_load_to_lds %0, %1, %2, %3"
                 :: "s"(g0), "s"(g1), "s"(gz), "s"(gz)
                 : "memory");
}

// Batched SpMM out[b] = M @ x[b], M shared across batch (COO).
// One workgroup owns the full LDS-resident output tile out[b, :, c0:c0+32]
// (1723*32 fp32 = 220KB -- only possible with CDNA5's 320KB LDS/WGP), so the
// scatter-add is entirely ds_add_f32 (zero global atomics). COO meta is
// double-buffered into LDS by the Tensor Data Mover (wave 0 issues 3
// tensor_load_to_lds per chunk; s_wait_tensorcnt + barrier publishes it),
// gathers run 8-deep per wave for MLP, and the tile is streamed out with
// async LDS->global b128 stores (no VGPR round-trip).
__global__ void ds_us_spmm_lds_kernel(const float* __restrict__ x,
                                      const float* __restrict__ vals,
                                      const int*   __restrict__ rows,
                                      const int*   __restrict__ cols,
                                      float*       __restrict__ out,
                                      int nnz, int n_in, int n_out)
{
    extern __shared__ float sh_out[];                 // [n_out * CW] accumulator
    __shared__ __align__(16) int   sh_rows[2][CHUNK];
    __shared__ __align__(16) int   sh_cols[2][CHUNK];
    __shared__ __align__(16) float sh_vals[2][CHUNK];

    const int tid  = threadIdx.x;
    const int lane = tid & 31;
    const int wave = tid >> 5;                        // 0..NWAVES-1
    const int b    = blockIdx.y;                      // batch index
    const int c0   = blockIdx.x * CW;                 // channel chunk base

    // per-thread gather base: x[b, c, c0+lane] = x[xbase + (size_t)c*CFEAT]
    const size_t xbase = (size_t)b * n_in * CFEAT + c0 + lane;

    // ---- zero the LDS accumulator tile (b128 stores) -----------------------
    const int tile_v4 = (n_out * CW) >> 2;            // divisible by 4
    v4f* sh_out4 = (v4f*)sh_out;
    for (int i = tid; i < tile_v4; i += NTHREADS)
        sh_out4[i] = (v4f)0.0f;

    // ---- TDM stage of one COO chunk into LDS (TENSORcnt-tracked) -----------
    // Executed by wave 0 only (tensor ops are per-wave DMA, EXEC-independent).
    auto stage = [&](int base, int buf) {
        int rem = nnz - base;
        if (rem < 0) rem = 0;
        int cb = (base < nnz) ? base : 0;             // keep tile start in-buffer
        tdm_load_1d(rows + cb, (uint32_t)(size_t)&sh_rows[buf][0], (uint32_t)rem);
        tdm_load_1d(cols + cb, (uint32_t)(size_t)&sh_cols[buf][0], (uint32_t)rem);
        tdm_load_1d(vals + cb, (uint32_t)(size_t)&sh_vals[buf][0], (uint32_t)rem);
    };

    if (wave == 0) stage(0, 0);                       // prologue: stage chunk 0

    const int nchunks = (nnz + CHUNK - 1) / CHUNK;
    for (int ck = 0; ck < nchunks; ++ck) {
        const int buf = ck & 1;
        if (wave == 0) {
            stage((ck + 1) * CHUNK, buf ^ 1);         // keep the pipeline full
            // 3 newest TDM ops may remain in flight; chunk ck's 3 are done
            // (tensor ops from one wave complete in order)
            __builtin_amdgcn_s_wait_tensorcnt(3);
        }
        __syncthreads();          // chunk-ck meta + (iter 0) zeroed tile visible

        int cnt = nnz - ck * CHUNK;
        if (cnt > CHUNK) cnt = CHUNK;

        // contiguous per-wave slice of the chunk, 8 nonzeros in flight
        const int jbeg = wave * SUB;
        int jn = cnt - jbeg;                           // entries for this wave
        if (jn > SUB) jn = SUB;

        int j = 0;
        for (; j + 8 <= jn; j += 8) {
            // broadcast meta for 8 nonzeros: two ds_load_b128 per array
            const v4i ra = *(const v4i*)&sh_rows[buf][jbeg + j];
            const v4i rb = *(const v4i*)&sh_rows[buf][jbeg + j + 4];
            const v4i ca = *(const v4i*)&sh_cols[buf][jbeg + j];
            const v4i cb = *(const v4i*)&sh_cols[buf][jbeg + j + 4];
            const v4f va = *(const v4f*)&sh_vals[buf][jbeg + j];
            const v4f vb = *(const v4f*)&sh_vals[buf][jbeg + j + 4];
            if (j + 16 <= jn) {                       // prefetch next group's row
                int cp = sh_cols[buf][jbeg + j + 8];
                __builtin_prefetch(&x[xbase + (size_t)cp * CFEAT], 0, 0);
            }
            // 8 independent coalesced 128B gathers (per-WGP MLP: 8 x 16 waves)
            const float x0 = x[xbase + (size_t)ca.x * CFEAT];
            const float x1 = x[xbase + (size_t)ca.y * CFEAT];
            const float x2 = x[xbase + (size_t)ca.z * CFEAT];
            const float x3 = x[xbase + (size_t)ca.w * CFEAT];
            const float x4 = x[xbase + (size_t)cb.x * CFEAT];
            const float x5 = x[xbase + (size_t)cb.y * CFEAT];
            const float x6 = x[xbase + (size_t)cb.z * CFEAT];
            const float x7 = x[xbase + (size_t)cb.w * CFEAT];
            // conflict-free LDS float atomic scatter-add (ds_add_f32)
            atomicAdd(&sh_out[ra.x * CW + lane], va.x * x0);
            atomicAdd(&sh_out[ra.y * CW + lane], va.y * x1);
            atomicAdd(&sh_out[ra.z * CW + lane], va.z * x2);
            atomicAdd(&sh_out[ra.w * CW + lane], va.w * x3);
            atomicAdd(&sh_out[rb.x * CW + lane], vb.x * x4);
            atomicAdd(&sh_out[rb.y * CW + lane], vb.y * x5);
            atomicAdd(&sh_out[rb.z * CW + lane], vb.z * x6);
            atomicAdd(&sh_out[rb.w * CW + lane], vb.w * x7);
        }
        for (; j < jn; ++j) {                          // scalar tail (last chunk)
            const int   r = sh_rows[buf][jbeg + j];
            const int   c = sh_cols[buf][jbeg + j];
            const float v = sh_vals[buf][jbeg + j];
            atomicAdd(&sh_out[r * CW + lane], v * x[xbase + (size_t)c * CFEAT]);
        }
        __syncthreads();          // all reads of buf done before it is restaged
    }

    // ---- stream the tile out: async LDS->global b128 stores ----------------
    // 4 rows per wave instruction: lanes 0-7 -> row r0, 8-15 -> r0+1, ...
    // ASYNCcnt-tracked; s_endpgm's implicit wait-idle guarantees completion.
    const int rsub = lane >> 3;                        // 0..3
    const int csub = (lane & 7) * 4;                   // 0,4,...,28
    const unsigned long long obase = (unsigned long long)(size_t)out;
    for (int r0 = wave * 4; r0 < n_out; r0 += NWAVES * 4) {
        const int rr = r0 + rsub;
        if (rr < n_out) {
            uint32_t loff = (uint32_t)(size_t)&sh_out[rr * CW + csub];
            uint32_t goff = (uint32_t)((((size_t)b * n_out + rr) * CFEAT
                                        + c0 + csub) * sizeof(float));
            asm volatile("global_store_async_from_lds_b128 %0, %1, %2"
                         :: "v"(goff), "v"(loff), "s"(obase)
                         : "memory");
        }
    }
}

extern "C" void kernel_launch(void* const* d_in, const int* in_sizes, int n_in_args,
                              void* d_out, int out_size, void* d_ws, size_t ws_size,
                              hipStream_t stream) {
    (void)d_ws; (void)ws_size; (void)n_in_args;

    const float* x    = (const float*)d_in[0];
    const float* vals = (const float*)d_in[1];
    const int*   rows = (const int*)d_in[2];
    const int*   cols = (const int*)d_in[3];
    // d_in[4] is n_out on device; derive dims host-side instead (deterministic)
    float* out = (float*)d_out;

    const int nnz   = in_sizes[1];
    const int n_in  = in_sizes[0] / (BATCH * CFEAT);   // 6890
    const int n_out = out_size    / (BATCH * CFEAT);   // 1723

    const size_t lds_bytes = (size_t)n_out * CW * sizeof(float);  // ~220.5 KB
    // idempotent, identical on every call (no static guards per harness rules)
    hipFuncSetAttribute((const void*)ds_us_spmm_lds_kernel,
                        hipFuncAttributeMaxDynamicSharedMemorySize,
                        (int)lds_bytes);

    dim3 grid(CFEAT / CW, BATCH);   // 8 x 32 = 256 workgroups (1 per WGP, 16 waves)
    dim3 block(NTHREADS);
    ds_us_spmm_lds_kernel<<<grid, block, lds_bytes, stream>>>(
        x, vals, rows, cols, out, nnz, n_in, n_out);
}